// MaskModule_14809047237010
// MI455X (gfx1250) — compile-verified
//
#include <hip/hip_runtime.h>

#define D_DIM 128
#define Q_DIM 100
#define QP_DIM 112   // Q padded to multiple of 16
#define C_DIM 20
#define B_DIM 4
#define N_PTS 262144
#define LN_EPS 1e-5f

typedef __attribute__((ext_vector_type(2))) float v2f;
typedef __attribute__((ext_vector_type(8))) float v8f;

// ---------------- kernel 1: LayerNorm over hidden dim ----------------
__global__ void ln_kernel(const float* __restrict__ x,
                          const float* __restrict__ gamma,
                          const float* __restrict__ beta,
                          float* __restrict__ qn) {
    int row  = blockIdx.x;          // 0..B*Q-1
    int lane = threadIdx.x;         // 0..31
    float4 v = ((const float4*)(x + (size_t)row * D_DIM))[lane];
    float s  = v.x + v.y + v.z + v.w;
    float ss = v.x * v.x + v.y * v.y + v.z * v.z + v.w * v.w;
    #pragma unroll
    for (int off = 16; off > 0; off >>= 1) {
        s  += __shfl_xor(s,  off, 32);
        ss += __shfl_xor(ss, off, 32);
    }
    float mu  = s * (1.0f / D_DIM);
    float var = ss * (1.0f / D_DIM) - mu * mu;
    float inv = rsqrtf(var + LN_EPS);
    float4 g = ((const float4*)gamma)[lane];
    float4 b = ((const float4*)beta)[lane];
    float4 o;
    o.x = (v.x - mu) * inv * g.x + b.x;
    o.y = (v.y - mu) * inv * g.y + b.y;
    o.z = (v.z - mu) * inv * g.z + b.z;
    o.w = (v.w - mu) * inv * g.w + b.w;
    ((float4*)(qn + (size_t)row * D_DIM))[lane] = o;
}

// ---------------- kernel 2a: h = relu(qn @ W1^T + b1) ----------------
__global__ void mlp1_kernel(const float* __restrict__ qn,
                            const float* __restrict__ W1,
                            const float* __restrict__ b1,
                            float* __restrict__ h) {
    __shared__ float row[D_DIM];
    int r = blockIdx.x;             // 0..B*Q-1
    int e = threadIdx.x;            // 0..127
    row[e] = qn[(size_t)r * D_DIM + e];
    __syncthreads();
    const float4* w = (const float4*)(W1 + (size_t)e * D_DIM);
    float acc = 0.0f;
    #pragma unroll
    for (int i = 0; i < D_DIM / 4; ++i) {
        float4 wv = w[i];
        acc += wv.x * row[4 * i + 0] + wv.y * row[4 * i + 1] +
               wv.z * row[4 * i + 2] + wv.w * row[4 * i + 3];
    }
    acc += b1[e];
    h[(size_t)r * D_DIM + e] = fmaxf(acc, 0.0f);
}

// ---------------- kernel 2b: mask_embed = h @ W2^T + b2 ----------------
// K-pair interleaved, transposed, padded layout:
//   me2[ ((b*(D/2) + k/2) * QP + q) * 2 + (k&1) ]
// so WMMA B operand (two consecutive K for a given q) is one aligned b64.
__global__ void mlp2_kernel(const float* __restrict__ h,
                            const float* __restrict__ W2,
                            const float* __restrict__ b2,
                            float* __restrict__ me2) {
    int b  = blockIdx.x / QP_DIM;
    int qp = blockIdx.x % QP_DIM;
    int e  = threadIdx.x;           // 0..127  (= output feature = K index)
    size_t dst = (((size_t)b * (D_DIM / 2) + (e >> 1)) * QP_DIM + qp) * 2 + (e & 1);
    if (qp >= Q_DIM) {              // uniform per block: safe early path
        me2[dst] = 0.0f;
        return;
    }
    __shared__ float row[D_DIM];
    row[e] = h[((size_t)b * Q_DIM + qp) * D_DIM + e];
    __syncthreads();
    const float4* w = (const float4*)(W2 + (size_t)e * D_DIM);
    float acc = 0.0f;
    #pragma unroll
    for (int i = 0; i < D_DIM / 4; ++i) {
        float4 wv = w[i];
        acc += wv.x * row[4 * i + 0] + wv.y * row[4 * i + 1] +
               wv.z * row[4 * i + 2] + wv.w * row[4 * i + 3];
    }
    me2[dst] = acc + b2[e];
}

// ---------------- kernel 2c: outputs_class = qn @ Wc^T + bc ----------------
__global__ void cls_kernel(const float* __restrict__ qn,
                           const float* __restrict__ Wc,
                           const float* __restrict__ bc,
                           float* __restrict__ outc) {
    __shared__ float row[D_DIM];
    int r    = blockIdx.x;          // 0..B*Q-1
    int lane = threadIdx.x;         // 0..31
    ((float4*)row)[lane] = ((const float4*)(qn + (size_t)r * D_DIM))[lane];
    __syncthreads();
    if (lane < C_DIM) {
        const float4* w = (const float4*)(Wc + (size_t)lane * D_DIM);
        float acc = 0.0f;
        #pragma unroll
        for (int i = 0; i < D_DIM / 4; ++i) {
            float4 wv = w[i];
            acc += wv.x * row[4 * i + 0] + wv.y * row[4 * i + 1] +
                   wv.z * row[4 * i + 2] + wv.w * row[4 * i + 3];
        }
        outc[(size_t)r * C_DIM + lane] = acc + bc[lane];
    }
}

// ---------------- kernel 3: out_masks via WMMA f32 16x16x4 ----------------
// block = 256 threads = 8 waves; each wave: 16 point rows x 112 query cols.
// mask_embed chunk (16 K-pairs x 112 q x 2) staged into LDS with
// global_load_async_to_lds_b128 (ASYNCcnt path), B operand read as ds_load_b64.
#define KC 32
#define ROWS 128
__global__ void __launch_bounds__(256)
mask_kernel(const float* __restrict__ pf,
            const int* __restrict__ bids,
            const float* __restrict__ me2,
            float* __restrict__ om) {
    __shared__ __attribute__((aligned(16))) float lme[KC * QP_DIM]; // 14336 B
    __shared__ int lbid[ROWS];

    int tid  = threadIdx.x;
    int row0 = blockIdx.x * ROWS;
    if (tid < ROWS) lbid[tid] = bids[row0 + tid];
    __syncthreads();
    int bFirst = lbid[0];
    int bLast  = lbid[ROWS - 1];     // batch_ids sorted -> batches in [bFirst,bLast]

    int wave = tid >> 5;
    int lane = tid & 31;
    int half = lane >> 4;            // 0: K pair {0,1}, 1: K pair {2,3}
    int idx  = lane & 15;            // M (for A) / N (for B,D)
    int rbase = row0 + wave * 16;
    const float* aptr = pf + (size_t)(rbase + idx) * D_DIM + 2 * half;
    unsigned lbase = (unsigned)(size_t)(void*)lme;   // LDS byte offset (low 32 bits)

    for (int b = bFirst; b <= bLast; ++b) {
        v8f acc[7] = {};
        const float* meb = me2 + (size_t)b * (D_DIM / 2) * QP_DIM * 2;

        for (int kc0 = 0; kc0 < D_DIM; kc0 += KC) {
            __syncthreads();         // previous chunk fully consumed
            // async copy: 3584 floats = 896 x b128, spread over 256 lanes
            {
                const float* src = meb + (size_t)(kc0 / 2) * QP_DIM * 2;
                for (int i = tid * 4; i < KC * QP_DIM; i += 1024) {
                    unsigned voff = (unsigned)(i * 4);   // byte offset
                    unsigned laddr = lbase + voff;
                    asm volatile("global_load_async_to_lds_b128 %0, %1, %2"
                                 :
                                 : "v"(laddr), "v"(voff), "s"(src)
                                 : "memory");
                }
            }
            asm volatile("s_wait_asynccnt 0x0" ::: "memory");
            __syncthreads();
            if (kc0 + KC < D_DIM)
                __builtin_prefetch(aptr + kc0 + KC, 0, 0);  // global_prefetch_b8

            #pragma unroll
            for (int kk = 0; kk < KC; kk += 4) {
                // A 16x4: lanes 0-15 hold K={kk,kk+1}, lanes 16-31 K={kk+2,kk+3}
                v2f a = *(const v2f*)(aptr + kc0 + kk);
                // B pairs for this k-step start at pair row kk/2 (+half)
                const v2f* brow = (const v2f*)&lme[((kk >> 1) + half) * QP_DIM * 2];
                #pragma unroll
                for (int t = 0; t < 7; ++t) {
                    int q = t * 16 + idx;
                    v2f bv = brow[q];          // one aligned ds_load_b64
                    acc[t] = __builtin_amdgcn_wmma_f32_16x16x4_f32(
                        /*neg_a=*/false, a, /*neg_b=*/false, bv,
                        /*c_mod=*/(short)0, acc[t],
                        /*reuse_a=*/false, /*reuse_b=*/false);
                }
            }
        }

        // store: VGPR r holds row M = r + 8*half, col N = idx (16x16 f32 D layout)
        #pragma unroll
        for (int t = 0; t < 7; ++t) {
            int q = t * 16 + idx;
            if (q < Q_DIM) {
                #pragma unroll
                for (int r = 0; r < 8; ++r) {
                    int lrow = wave * 16 + r + 8 * half;
                    if (lbid[lrow] == b)
                        om[(size_t)(row0 + lrow) * Q_DIM + q] = acc[t][r];
                }
            }
        }
    }
}

extern "C" void kernel_launch(void* const* d_in, const int* in_sizes, int n_in,
                              void* d_out, int out_size, void* d_ws, size_t ws_size,
                              hipStream_t stream) {
    const float* query = (const float*)d_in[0];
    const float* pf    = (const float*)d_in[1];
    const int*   bids  = (const int*)d_in[2];
    const float* gamma = (const float*)d_in[3];
    const float* beta  = (const float*)d_in[4];
    const float* W1    = (const float*)d_in[5];
    const float* b1    = (const float*)d_in[6];
    const float* W2    = (const float*)d_in[7];
    const float* b2    = (const float*)d_in[8];
    const float* Wc    = (const float*)d_in[9];
    const float* bc    = (const float*)d_in[10];

    float* outc = (float*)d_out;                       // [B,Q,C] = 8000 floats
    float* om   = outc + B_DIM * Q_DIM * C_DIM;        // [N,Q]

    float* ws = (float*)d_ws;
    float* qn = ws;                                    // 51200
    float* h  = ws + (size_t)B_DIM * Q_DIM * D_DIM;    // 51200
    float* me = h  + (size_t)B_DIM * Q_DIM * D_DIM;    // B*(D/2)*QP*2 = 57344

    ln_kernel  <<<B_DIM * Q_DIM, 32,  0, stream>>>(query, gamma, beta, qn);
    mlp1_kernel<<<B_DIM * Q_DIM, 128, 0, stream>>>(qn, W1, b1, h);
    mlp2_kernel<<<B_DIM * QP_DIM, 128, 0, stream>>>(h, W2, b2, me);
    cls_kernel <<<B_DIM * Q_DIM, 32,  0, stream>>>(qn, Wc, bc, outc);
    mask_kernel<<<N_PTS / ROWS, 256, 0, stream>>>(pf, bids, me, om);
}